// GELU12_17566416240705
// MI455X (gfx1250) — compile-verified
//
#include <hip/hip_runtime.h>
#include <math.h>

typedef __attribute__((ext_vector_type(2))) float v2f;
typedef __attribute__((ext_vector_type(4))) float v4f;
typedef __attribute__((ext_vector_type(8))) float v8f;

#define EPSN 1e-8f

// ---------------------------------------------------------------------------
// Kernel 1: normalize prototypes into d_ws, padded with a zero row (row 8)
// so WMMA B-lanes 8..15 can all read the zero row. Grid = 9 blocks x 256 thr.
// ---------------------------------------------------------------------------
__global__ __launch_bounds__(256) void proto_prep_kernel(
    const float* __restrict__ P, float* __restrict__ Pn, int D)
{
    __shared__ float red[256];
    const int row = blockIdx.x;
    const int tid = threadIdx.x;

    if (row == 8) {  // zero pad row
        for (int d = tid; d < D; d += 256) Pn[(size_t)8 * D + d] = 0.0f;
        return;
    }

    const float* p = P + (size_t)row * D;
    float s = 0.0f;
    for (int d = tid; d < D; d += 256) { float v = p[d]; s = fmaf(v, v, s); }
    red[tid] = s;
    __syncthreads();
    for (int off = 128; off > 0; off >>= 1) {
        if (tid < off) red[tid] += red[tid + off];
        __syncthreads();
    }
    const float inv = 1.0f / fmaxf(sqrtf(red[0]), EPSN);
    for (int d = tid; d < D; d += 256) Pn[(size_t)row * D + d] = p[d] * inv;
}

__device__ __forceinline__ float gelu_tanh(float x) {
    // 0.5*x*(1 + tanh(sqrt(2/pi) * x * (1 + 0.044715*x^2)))
    const float c = 0.7978845608028654f;
    float inner = c * x * fmaf(0.044715f * x, x, 1.0f);
    return 0.5f * x * (1.0f + tanhf(inner));
}

// ---------------------------------------------------------------------------
// Kernel 2: one wave per 16-row tile.
//   Pass 1: stream tile (RT loads -> retained in L2), WMMA f32 16x16x4
//           accumulates sims (C = X_tile * P^T), VALU accumulates sumsq.
//   Cross-lane: fold norms, clamp, min over K=8 protos, compute per-row scale.
//   Pass 2: re-read tile (NT loads: L2-hit then evict-first) and NT stores
//           (rinse output through L2 so it never displaces live x tiles).
// Block = 256 threads = 8 waves = 128 rows. LDS holds padded prototypes (72KB).
// ---------------------------------------------------------------------------
__global__ __launch_bounds__(256) void novelty_gelu_kernel(
    const float* __restrict__ x, const float* __restrict__ Pn,
    const float* __restrict__ log_tau, const float* __restrict__ log_blend,
    float* __restrict__ out, int D)
{
    extern __shared__ float sP[];  // 9 * D floats

    const int tid  = threadIdx.x;
    const int lane = tid & 31;
    const int wave = tid >> 5;

    // Stage padded normalized prototypes into LDS (once; waves independent after).
    const int nP = 9 * D;
    for (int i = tid; i < nP; i += 256) sP[i] = Pn[i];
    __syncthreads();

    const float tau   = expf(log_tau[0]);
    const float alpha = 1.0f / (1.0f + expf(-log_blend[0]));

    const size_t rowBase = (size_t)blockIdx.x * 128 + (size_t)wave * 16;

    // WMMA A-layout addressing: lanes 0-15 hold K=0,1 ; lanes 16-31 hold K=2,3.
    const int r    = lane & 15;        // row within tile (A) / proto col N (B)
    const int koff = (lane >> 4) * 2;  // K sub-offset within the 4-wide chunk
    const float* ax = x + (rowBase + r) * (size_t)D + koff;
    const int   prow = (r < 8) ? r : 8;             // pad lanes read zero row
    const float* bx  = sP + (size_t)prow * D + koff;

    v8f   c  = {};     // sims accumulator: 16x16 C tile
    float sq = 0.0f;   // partial sum-of-squares for row r (half of K-space)

#pragma unroll 8
    for (int d0 = 0; d0 < D; d0 += 4) {
        v2f a = *(const v2f*)(ax + d0);   // A: x[row][d0+koff .. +1] (RT: keep in L2)
        v2f b = *(const v2f*)(bx + d0);   // B: p[n][d0+koff .. +1] from LDS
        c = __builtin_amdgcn_wmma_f32_16x16x4_f32(
                false, a, false, b, (short)0, c, false, false);
        sq = fmaf(a.x, a.x, fmaf(a.y, a.y, sq));
    }

    // Row norm: lanes r and r+16 jointly cover the row -> fold across xor 16.
    sq += __shfl_xor(sq, 16, 32);
    const float inv = 1.0f / fmaxf(sqrtf(sq), EPSN);  // matches x/max(||x||,eps)

    // C layout: VGPR j, lane L -> sims[M][N], M = j + 8*(L>>4), N = L&15.
    const int g = lane >> 4;
    float dmin[8];
#pragma unroll
    for (int j = 0; j < 8; ++j) {
        const float invM = __shfl(inv, j + 8 * g, 32);  // norm of row M
        float sim = c[j] * invM;
        sim = fminf(fmaxf(sim, -1.0f), 1.0f);
        float dist = 1.0f - sim;
        if (r >= 8) dist = 3.0e38f;                     // pad protos excluded
        // min over the 8 valid proto lanes (stays inside aligned groups of 8)
        dist = fminf(dist, __shfl_xor(dist, 4, 32));
        dist = fminf(dist, __shfl_xor(dist, 2, 32));
        dist = fminf(dist, __shfl_xor(dist, 1, 32));
        dmin[j] = dist;  // row (j + 8g)'s min-dist lives in lanes 16g..16g+7
    }

    // Gather my row's min-dist: row r = (r&7) + 8*(r>>3).
    float md = 0.0f;
    const int src = (r >> 3) * 16;
#pragma unroll
    for (int j = 0; j < 8; ++j) {
        const float v = __shfl(dmin[j], src, 32);
        md = ((r & 7) == j) ? v : md;
    }
    md = fminf(fmaxf(md, 0.0f), 2.0f);
    const float novelty = 1.0f - expf(-tau * md);
    float scale = fmaf(alpha, novelty, 1.0f - alpha);
    scale = fminf(fmaxf(scale, 0.1f), 10.0f);  // per-row scale, held by lane r

    // Pass 2: tile is hot in L2; coalesced float4 walk over the 16 rows.
    // NT loads (last use of x) + NT stores (don't pollute L2 with output).
    const int nv4 = D >> 2;
    for (int rr = 0; rr < 16; ++rr) {
        const float s = __shfl(scale, rr, 32);
        const v4f* xin = (const v4f*)(x   + (rowBase + rr) * (size_t)D);
        v4f*       o   = (v4f*)      (out + (rowBase + rr) * (size_t)D);
        for (int i = lane; i < nv4; i += 32) {
            v4f v = __builtin_nontemporal_load(xin + i);
            v4f w;
            w.x = gelu_tanh(v.x * s);
            w.y = gelu_tanh(v.y * s);
            w.z = gelu_tanh(v.z * s);
            w.w = gelu_tanh(v.w * s);
            __builtin_nontemporal_store(w, o + i);
        }
    }
}

extern "C" void kernel_launch(void* const* d_in, const int* in_sizes, int n_in,
                              void* d_out, int out_size, void* d_ws, size_t ws_size,
                              hipStream_t stream) {
    const float* x         = (const float*)d_in[0];
    const float* P         = (const float*)d_in[1];
    const float* log_tau   = (const float*)d_in[2];
    const float* log_blend = (const float*)d_in[3];
    float* out = (float*)d_out;

    const int K = 8;
    const int D = in_sizes[1] / K;                         // 2048
    const long long rows = (long long)in_sizes[0] / D;     // 32768

    float* Pn = (float*)d_ws;  // 9*D floats (normalized protos + zero pad row)

    proto_prep_kernel<<<9, 256, 0, stream>>>(P, Pn, D);

    const int rowsPerBlock = 128;                          // 8 waves x 16 rows
    const int nBlocks = (int)(rows / rowsPerBlock);        // 256
    const size_t shmem = (size_t)9 * D * sizeof(float);    // 72 KB
    novelty_gelu_kernel<<<nBlocks, 256, shmem, stream>>>(
        x, Pn, log_tau, log_blend, out, D);
}